// RoIAlign_31877247270970
// MI455X (gfx1250) — compile-verified
//
#include <hip/hip_runtime.h>
#include <hip/hip_bf16.h>

// ---------------- problem constants (from reference) ----------------
#define OUT_HW 7
#define SS     2           // sampling ratio per axis
#define NB     4
#define CC     256
#define HH     200
#define WW     200
#define RR     512
#define SCALE  0.25f

#define WIN    28          // per-axis tap window (roi span <= 26 incl. taps)
#define KSTEPS (WIN / 4)   // 7 wmma k-steps of K=4
#define CBLK   64          // channels processed per LDS slab
#define CTILES (CBLK / 16) // 4 wmma N-tiles per slab
#define TPAD   (CBLK + 1)  // pad inner dim: (2*7*65) % 64 != 0 -> no bank conflict

typedef float v2f __attribute__((ext_vector_type(2)));
typedef float v8f __attribute__((ext_vector_type(8)));

__device__ __forceinline__ int imin(int a, int b) { return a < b ? a : b; }
__device__ __forceinline__ int imax(int a, int b) { return a > b ? a : b; }

// ---------------- NCHW -> NHWC transpose (coalesced both sides via LDS tile) ----
__global__ __launch_bounds__(256) void nchw_to_nhwc(const float* __restrict__ in,
                                                    float* __restrict__ out) {
    __shared__ float tile[32][33];
    const int w0 = blockIdx.x * 32;
    const int c0 = blockIdx.y * 32;
    const int nh = blockIdx.z;
    const int n = nh / HH, h = nh % HH;

    for (int i = threadIdx.y; i < 32; i += 8) {
        const int c = c0 + i;
        const int w = w0 + threadIdx.x;
        tile[i][threadIdx.x] = (w < WW) ? in[(((size_t)n * CC + c) * HH + h) * WW + w] : 0.0f;
    }
    __syncthreads();
    for (int i = threadIdx.y; i < 32; i += 8) {
        const int w = w0 + i;
        const int c = c0 + threadIdx.x;
        if (w < WW)
            out[(((size_t)n * HH + h) * WW + w) * CC + c] = tile[threadIdx.x][i];
    }
}

// ---------------- WMMA RoIAlign: one workgroup (8 wave32) per roi -------------
__global__ __launch_bounds__(256) void roialign_wmma(const float* __restrict__ nhwc,
                                                     const float* __restrict__ rois,
                                                     float* __restrict__ out) {
    __shared__ float Ax[16][WIN];                 // x weights: rows = pw (7 used)
    __shared__ float Ay[16][WIN];                 // y weights: rows = ph (7 used)
    __shared__ float T[WIN][OUT_HW][TPAD];        // stage-1 result, 64-ch slab
    __shared__ int   geom[4];                     // n, xbase, ybase

    const int r    = blockIdx.x;
    const int tid  = threadIdx.x;
    const int lane = tid & 31;
    const int wave = tid >> 5;

    // zero weight tables
    for (int i = tid; i < 16 * WIN; i += 256) {
        (&Ax[0][0])[i] = 0.0f;
        (&Ay[0][0])[i] = 0.0f;
    }
    __syncthreads();

    if (tid == 0) {
        const float* rp = rois + (size_t)r * 5;
        const int   n  = (int)rp[0];
        const float x1 = rp[1] * SCALE, y1 = rp[2] * SCALE;
        const float x2 = rp[3] * SCALE, y2 = rp[4] * SCALE;
        const float rw = fmaxf(x2 - x1, 1.0f), rh = fmaxf(y2 - y1, 1.0f);
        const float bw = rw * (1.0f / OUT_HW), bh = rh * (1.0f / OUT_HW);

        // window bases from first (smallest) sample coordinate
        const float xc0 = fminf(fmaxf(x1 + 0.25f * bw, 0.0f), (float)(WW - 1));
        const float yc0 = fminf(fmaxf(y1 + 0.25f * bh, 0.0f), (float)(HH - 1));
        const int xbase = imin((int)floorf(xc0), WW - WIN);
        const int ybase = imin((int)floorf(yc0), HH - WIN);
        geom[0] = n; geom[1] = xbase; geom[2] = ybase;

        for (int i = 0; i < OUT_HW * SS; ++i) {
            const float fi = ((float)i + 0.5f) * 0.5f;   // (i+0.5)/S
            const int   p  = i >> 1;                     // output bin, pool folded in
            // ---- X axis ----
            {
                const float xs = x1 + fi * bw;
                const float vx = (xs >= -1.0f && xs <= (float)WW) ? 1.0f : 0.0f;
                const float xc = fminf(fmaxf(xs, 0.0f), (float)(WW - 1));
                const int   xl = (int)floorf(xc);
                const int   xh = imin(xl + 1, WW - 1);
                const float lx = xc - (float)xl;
                const int ul = imin(imax(xl - xbase, 0), WIN - 1);
                const int uh = imin(imax(xh - xbase, 0), WIN - 1);
                Ax[p][ul] += 0.5f * (1.0f - lx) * vx;    // 0.5 = sqrt of 1/(S*S)
                Ax[p][uh] += 0.5f * lx * vx;
            }
            // ---- Y axis ----
            {
                const float ys = y1 + fi * bh;
                const float vy = (ys >= -1.0f && ys <= (float)HH) ? 1.0f : 0.0f;
                const float yc = fminf(fmaxf(ys, 0.0f), (float)(HH - 1));
                const int   yl = (int)floorf(yc);
                const int   yh = imin(yl + 1, HH - 1);
                const float ly = yc - (float)yl;
                const int ul = imin(imax(yl - ybase, 0), WIN - 1);
                const int uh = imin(imax(yh - ybase, 0), WIN - 1);
                Ay[p][ul] += 0.5f * (1.0f - ly) * vy;
                Ay[p][uh] += 0.5f * ly * vy;
            }
        }
    }
    __syncthreads();

    const int n     = geom[0];
    const int xbase = geom[1];
    const int ybase = geom[2];

    const int arow = lane & 15;          // fragment row (M) / col (N) index
    const int kk   = (lane >> 4) << 1;   // 0 for lanes 0-15, 2 for lanes 16-31

    for (int cb = 0; cb < CC; cb += CBLK) {
        // ---------- stage 1: T[v][pw][c] = sum_u Ax[pw][u] * feat[ybase+v][xbase+u][c]
        for (int chain = wave; chain < WIN * CTILES; chain += 8) {
            const int v  = chain >> 2;
            const int ct = chain & 3;
            const float* fb = nhwc +
                (((size_t)n * HH + (ybase + v)) * WW + xbase) * CC +
                cb + ct * 16 + arow;
            v8f acc = {0.0f, 0.0f, 0.0f, 0.0f, 0.0f, 0.0f, 0.0f, 0.0f};
#pragma unroll
            for (int j = 0; j < KSTEPS; ++j) {
                const int k0 = j * 4;
                v2f a, b;
                a.x = Ax[arow][k0 + kk];
                a.y = Ax[arow][k0 + kk + 1];
                b.x = fb[(size_t)(k0 + kk) * CC];
                b.y = fb[(size_t)(k0 + kk + 1) * CC];
                acc = __builtin_amdgcn_wmma_f32_16x16x4_f32(
                    false, a, false, b, (short)0, acc, false, false);
            }
            if (lane < 16) {
#pragma unroll
                for (int i = 0; i < OUT_HW; ++i)
                    T[v][i][ct * 16 + lane] = acc[i];   // rows 7..15 are zero-weight
            }
        }
        __syncthreads();

        // ---------- stage 2: out[ph][pw][c] = sum_v Ay[ph][v] * T[v][pw][c]
        for (int chain = wave; chain < OUT_HW * CTILES; chain += 8) {
            const int pw = chain >> 2;
            const int ct = chain & 3;
            v8f acc = {0.0f, 0.0f, 0.0f, 0.0f, 0.0f, 0.0f, 0.0f, 0.0f};
#pragma unroll
            for (int j = 0; j < KSTEPS; ++j) {
                const int k0 = j * 4;
                v2f a, b;
                a.x = Ay[arow][k0 + kk];
                a.y = Ay[arow][k0 + kk + 1];
                b.x = T[k0 + kk][pw][ct * 16 + arow];
                b.y = T[k0 + kk + 1][pw][ct * 16 + arow];
                acc = __builtin_amdgcn_wmma_f32_16x16x4_f32(
                    false, a, false, b, (short)0, acc, false, false);
            }
            if (lane < 16) {
                const int c = cb + ct * 16 + lane;
#pragma unroll
                for (int i = 0; i < OUT_HW; ++i)
                    out[(((size_t)r * CC + c) * OUT_HW + i) * OUT_HW + pw] = acc[i];
            }
        }
        __syncthreads();   // T is reused next channel slab
    }
}

// ---------------- fallback: direct NCHW gather (if ws can't hold NHWC) --------
__global__ __launch_bounds__(256) void roialign_direct(const float* __restrict__ feat,
                                                       const float* __restrict__ rois,
                                                       float* __restrict__ out,
                                                       int total) {
    const int o = blockIdx.x * 256 + threadIdx.x;
    if (o >= total) return;
    const int pw = o % OUT_HW;
    const int ph = (o / OUT_HW) % OUT_HW;
    const int c  = (o / (OUT_HW * OUT_HW)) % CC;
    const int r  = o / (OUT_HW * OUT_HW * CC);

    const float* rp = rois + (size_t)r * 5;
    const int   n  = (int)rp[0];
    const float x1 = rp[1] * SCALE, y1 = rp[2] * SCALE;
    const float x2 = rp[3] * SCALE, y2 = rp[4] * SCALE;
    const float rw = fmaxf(x2 - x1, 1.0f), rh = fmaxf(y2 - y1, 1.0f);
    const float bw = rw * (1.0f / OUT_HW), bh = rh * (1.0f / OUT_HW);
    const float* fp = feat + ((size_t)n * CC + c) * HH * WW;

    float acc = 0.0f;
#pragma unroll
    for (int sy = 0; sy < SS; ++sy) {
        const float ys = y1 + (((float)(ph * SS + sy) + 0.5f) * 0.5f) * bh;
        const float vy = (ys >= -1.0f && ys <= (float)HH) ? 1.0f : 0.0f;
        const float yc = fminf(fmaxf(ys, 0.0f), (float)(HH - 1));
        const int   yl = (int)floorf(yc);
        const int   yh = imin(yl + 1, HH - 1);
        const float ly = yc - (float)yl;
#pragma unroll
        for (int sx = 0; sx < SS; ++sx) {
            const float xs = x1 + (((float)(pw * SS + sx) + 0.5f) * 0.5f) * bw;
            const float vx = (xs >= -1.0f && xs <= (float)WW) ? 1.0f : 0.0f;
            const float xc = fminf(fmaxf(xs, 0.0f), (float)(WW - 1));
            const int   xl = (int)floorf(xc);
            const int   xh = imin(xl + 1, WW - 1);
            const float lx = xc - (float)xl;
            const float v00 = fp[yl * WW + xl], v01 = fp[yl * WW + xh];
            const float v10 = fp[yh * WW + xl], v11 = fp[yh * WW + xh];
            acc += vy * vx * ((1.0f - ly) * ((1.0f - lx) * v00 + lx * v01) +
                              ly * ((1.0f - lx) * v10 + lx * v11));
        }
    }
    out[o] = acc * 0.25f;
}

// ---------------- launch ------------------------------------------------------
extern "C" void kernel_launch(void* const* d_in, const int* in_sizes, int n_in,
                              void* d_out, int out_size, void* d_ws, size_t ws_size,
                              hipStream_t stream) {
    const float* features = (const float*)d_in[0];
    const float* rois     = (const float*)d_in[1];
    float*       out      = (float*)d_out;

    const size_t need = (size_t)NB * CC * HH * WW * sizeof(float);
    if (d_ws != nullptr && ws_size >= need) {
        float* nhwc = (float*)d_ws;
        dim3 tg((WW + 31) / 32, CC / 32, NB * HH);
        dim3 tb(32, 8);
        nchw_to_nhwc<<<tg, tb, 0, stream>>>(features, nhwc);
        roialign_wmma<<<RR, 256, 0, stream>>>(nhwc, rois, out);
    } else {
        const int total = RR * CC * OUT_HW * OUT_HW;
        roialign_direct<<<(total + 255) / 256, 256, 0, stream>>>(features, rois, out, total);
    }
}